// GEVBeamformer_31971736551857
// MI455X (gfx1250) — compile-verified
//
#include <hip/hip_runtime.h>
#include <math.h>

// Problem constants (from reference): B=8, M=8, F=513, T=1000
#define B_ 8
#define M_ 8
#define F_ 513
#define T_ 1000

typedef __attribute__((ext_vector_type(2))) float v2f;
typedef __attribute__((ext_vector_type(8))) float v8f;

// ---------------------------------------------------------------------------
// 8x8 fp32 matmul  P = X * Y  (or X * Y^T) via V_WMMA_F32_16X16X4_F32.
// All matrices live in zero-padded 16x16 LDS tiles, so every lane's fragment
// load and D write-back is UNCONDITIONAL (no exec-mask divergence around the
// WMMA): rows/cols 8..15 are zero by invariant and stay zero because the
// zero A-rows / B-cols produce zero D entries there.
// Layouts (ISA 7.12.2):
//   A 16x4 : lane L<16 -> (M=L, K=ko+0/1); lane>=16 -> (M=L-16, K=ko+2/3)
//   B 4x16 : lane L<16 -> (K=ko+0/1, N=L); lane>=16 -> (K=ko+2/3, N=L-16)
//   C/D    : v[r] = (M=r, N=lane) lanes 0-15 ; (M=r+8, N=lane-16) lanes 16-31
// ---------------------------------------------------------------------------
__device__ __forceinline__ void mm8x8_wmma(const float* __restrict__ X,
                                           const float* __restrict__ Y,
                                           float* __restrict__ P,
                                           int lane, bool transY) {
  v8f acc = {};
  const int n = lane & 15;
  const int hi = lane >> 4;
#pragma unroll
  for (int ko = 0; ko < 8; ko += 4) {
    const int kb = ko + (hi << 1);  // lanes>=16 handle K = ko+2, ko+3
    v2f a, bb;
    a[0] = X[n * 16 + kb];          // adjacent -> one ds_load_b64
    a[1] = X[n * 16 + kb + 1];
    if (transY) {
      bb[0] = Y[n * 16 + kb];
      bb[1] = Y[n * 16 + kb + 1];
    } else {
      bb[0] = Y[kb * 16 + n];
      bb[1] = Y[(kb + 1) * 16 + n];
    }
    // 8 args: (neg_a, A, neg_b, B, c_mod, C, reuse_a, reuse_b)
    acc = __builtin_amdgcn_wmma_f32_16x16x4_f32(false, a, false, bb,
                                                (short)0, acc, false, false);
  }
  // Unconditional full-tile write-back; rows/cols >=8 write zeros,
  // preserving the tile's zero-padding invariant.
  const int row0 = hi << 3;  // 0 or 8
#pragma unroll
  for (int r = 0; r < 8; ++r) P[(row0 + r) * 16 + n] = acc[r];
}

// ---------------------------------------------------------------------------
// Kernel 1: one wave32 per (b,f). Produces w[b][m][f] in d_ws.
//   1. condition noise SCM (gamma ridge)     2. Cholesky (lane 0)
//   3. L^-1 columns (lanes 0..7)             4. C = L^-1 A L^-T (2 WMMA mm)
//   5. C <- C^32 via 5 WMMA squarings with max-abs rescaling
//   6. v = C^32*ones ; e = L^-T v ; w = e/||e||
// ---------------------------------------------------------------------------
__global__ void __launch_bounds__(32)
gev_solve_kernel(const float* __restrict__ tgt,   // (B, M, M, F)
                 const float* __restrict__ noz,   // (B, M, M, F)
                 float* __restrict__ w) {         // (B, M, F)
  const int bf = blockIdx.x;
  const int b = bf / F_;
  const int f = bf - b * F_;
  const int lane = threadIdx.x;

  // Five 16x16 zero-padded tiles: N(->L), A, Linv, T, C
  __shared__ float sMem[5 * 256];
  float* const sN  = sMem + 0 * 256;
  float* const sA  = sMem + 1 * 256;
  float* const sLi = sMem + 2 * 256;
  float* const sT  = sMem + 3 * 256;
  float* const sC  = sMem + 4 * 256;
  __shared__ float sv[16];
  __shared__ float sred;

  // Zero-fill all tiles once (establishes the padding invariant).
  for (int e = lane; e < 5 * 256; e += 32) sMem[e] = 0.0f;
  __syncthreads();

  // Load both 8x8 SCMs into the top-left of their tiles.
  for (int e = lane; e < 64; e += 32) {
    const int i = e >> 3, j = e & 7;
    const size_t off = (size_t)(b * 64 + e) * F_ + f;
    sA[i * 16 + j] = tgt[off];
    sN[i * 16 + j] = noz[off];
  }
  __syncthreads();

  // ---- condition covariance + Cholesky (serial on lane 0; ~120 flops) ----
  if (lane == 0) {
    const float gamma = 1e-6f;
    float tr = 0.0f;
    for (int i = 0; i < 8; ++i) tr += sN[i * 17];
    const float ridge = gamma * tr / 8.0f;
    for (int i = 0; i < 8; ++i) sN[i * 17] += ridge;
    const float inv1pg = 1.0f / (1.0f + gamma);
    for (int i = 0; i < 8; ++i)
      for (int j = 0; j < 8; ++j) sN[i * 16 + j] *= inv1pg;
    // right-looking Cholesky; lower triangle of sN becomes L
    for (int k = 0; k < 8; ++k) {
      const float d = sqrtf(sN[k * 17]);
      sN[k * 17] = d;
      const float id = 1.0f / d;
      for (int i = k + 1; i < 8; ++i) sN[i * 16 + k] *= id;
      for (int j = k + 1; j < 8; ++j)
        for (int i = j; i < 8; ++i)
          sN[i * 16 + j] -= sN[i * 16 + k] * sN[j * 16 + k];
    }
  }
  __syncthreads();

  // ---- triangular inverse: lane j computes column j by forward subst ----
  if (lane < 8) {
    const int j = lane;
    sLi[j * 17] = 1.0f / sN[j * 17];
    for (int i = j + 1; i < 8; ++i) {
      float s = 0.0f;
      for (int k = j; k < i; ++k) s += sN[i * 16 + k] * sLi[k * 16 + j];
      sLi[i * 16 + j] = -s / sN[i * 17];
    }
  }
  __syncthreads();

  // ---- whitened matrix C = L^-1 * A * L^-T (WMMA) ----
  mm8x8_wmma(sLi, sA, sT, lane, /*transY=*/false);  // T = L^-1 A
  __syncthreads();
  mm8x8_wmma(sT, sLi, sC, lane, /*transY=*/true);   // C = T * (L^-1)^T
  __syncthreads();

  // ---- principal eigenvector via 5 WMMA squarings: C^32 ----
#pragma unroll
  for (int it = 0; it < 5; ++it) {
    if (lane == 0) {
      float mx = 1e-30f;
      for (int i = 0; i < 8; ++i)
        for (int j = 0; j < 8; ++j) mx = fmaxf(mx, fabsf(sC[i * 16 + j]));
      sred = 1.0f / mx;
    }
    __syncthreads();
    // Rescale whole tile (padding stays zero): sT = sC / max|C|
    for (int e = lane; e < 256; e += 32) sT[e] = sC[e] * sred;
    __syncthreads();
    mm8x8_wmma(sT, sT, sC, lane, /*transY=*/false);  // C <- C^2
    __syncthreads();
  }

  // ---- v = C^32 * ones  ->  e = L^-T v  ->  normalize ----
  if (lane < 8) {
    float s = 0.0f;
    for (int k = 0; k < 8; ++k) s += sC[lane * 16 + k];
    sv[lane] = s;
  }
  __syncthreads();
  if (lane < 8) {
    float s = 0.0f;
    for (int k = 0; k < 8; ++k) s += sLi[k * 16 + lane] * sv[k];  // (L^-1)^T v
    sv[8 + lane] = s;
  }
  __syncthreads();
  if (lane == 0) {
    float nrm = 0.0f;
    for (int i = 0; i < 8; ++i) nrm += sv[8 + i] * sv[8 + i];
    sred = 1.0f / sqrtf(fmaxf(nrm, 1e-30f));
  }
  __syncthreads();
  if (lane < 8) w[((size_t)b * 8 + lane) * F_ + f] = sv[8 + lane] * sred;
}

// ---------------------------------------------------------------------------
// Kernel 2: out[b,f,t] = sum_m w[b,m,f] * mix[b,m,f,t]
// Pure HBM stream: 131 MB in + 16.4 MB out ~= 6.3 us at 23.3 TB/s.
// float4 (global_load_b128) over contiguous T; one block per (b,f).
// ---------------------------------------------------------------------------
__global__ void __launch_bounds__(256)
gev_apply_kernel(const float* __restrict__ mix,   // (B, M, F, T)
                 const float* __restrict__ w,     // (B, M, F)
                 float* __restrict__ out) {       // (B, F, T)
  const int bf = blockIdx.x;
  const int b = bf / F_;
  const int f = bf - b * F_;
  const int tid = threadIdx.x;

  __shared__ float sw[8];
  if (tid < 8) sw[tid] = w[((size_t)b * 8 + tid) * F_ + f];
  __syncthreads();

  if (tid < T_ / 4) {  // 250 float4 chunks cover T=1000
    const size_t t0 = (size_t)tid * 4;
    float4 acc = make_float4(0.f, 0.f, 0.f, 0.f);
#pragma unroll
    for (int m = 0; m < 8; ++m) {
      const size_t off = ((size_t)(b * 8 + m) * F_ + f) * T_ + t0;  // 16B aligned
      const float4 v = *(const float4*)(mix + off);
      const float wm = sw[m];
      acc.x = fmaf(wm, v.x, acc.x);
      acc.y = fmaf(wm, v.y, acc.y);
      acc.z = fmaf(wm, v.z, acc.z);
      acc.w = fmaf(wm, v.w, acc.w);
    }
    *(float4*)(out + ((size_t)b * F_ + f) * T_ + t0) = acc;
  }
}

extern "C" void kernel_launch(void* const* d_in, const int* in_sizes, int n_in,
                              void* d_out, int out_size, void* d_ws, size_t ws_size,
                              hipStream_t stream) {
  const float* mix = (const float*)d_in[0];   // (B, M, F, T)
  const float* tgt = (const float*)d_in[1];   // (B, M, M, F)
  const float* noz = (const float*)d_in[2];   // (B, M, M, F)
  float* out = (float*)d_out;                 // (B, F, T)
  float* w   = (float*)d_ws;                  // (B, M, F) = 131,328 bytes scratch

  // Phase 1: 4104 independent 8x8 GEV solves, one wave32 each (WMMA path)
  gev_solve_kernel<<<dim3(B_ * F_), dim3(32), 0, stream>>>(tgt, noz, w);
  // Phase 2: bandwidth-bound beamformer application
  gev_apply_kernel<<<dim3(B_ * F_), dim3(256), 0, stream>>>(mix, w, out);
}